// RegionLoss_72851235275376
// MI455X (gfx1250) — compile-verified
//
#include <hip/hip_runtime.h>
#include <math.h>

// Problem dims (fixed by reference setup_inputs)
#define NB 32
#define NA 5
#define NC 20
#define NH 64
#define NW 64
#define TBOX 50
#define REC 16            // floats per GT record in workspace
#define PLANE (NH*NW)     // 4096
#define BLKS_PER_PLANE (PLANE/256)   // 16
#define OBJ_SCALE 5.0f
// inter/union > 0.6  <=>  inter > (0.6/1.6)*(parea+garea)
#define SIL_FRAC 0.375f   // 0.6 / (1 + 0.6)

typedef float v2f __attribute__((ext_vector_type(2)));
typedef float v8f __attribute__((ext_vector_type(8)));

// fast sigmoid: v_exp_f32 + v_rcp_f32 (hardware transcendentals, ~1 ulp each)
__device__ __forceinline__ float sigmoidf_(float v) {
    return __builtin_amdgcn_rcpf(1.0f + __expf(-v));
}

// Exact full-wave32 sum using V_WMMA_F32_16X16X4_F32.
// A[m][0] = partial(lane m), A[m][2] = partial(lane m+16), B = ones:
// D[m][n] = p_m + p_{m+16}. Lane l holds D rows {0..7} (l<16) or {8..15} (l>=16)
// of column l%16, so sum(D VGPRs) + shfl_xor(16) = total over 32 lanes (exact f32 adds).
__device__ __forceinline__ float wave32_sum_wmma(float partial) {
    v2f A; A[0] = partial; A[1] = 0.0f;   // lanes 0-15: K0; lanes 16-31: K2 (K1/K3 zero)
    v2f B; B[0] = 1.0f;   B[1] = 1.0f;    // all 64 B elements = 1.0
    v8f C = {0.f,0.f,0.f,0.f,0.f,0.f,0.f,0.f};
    v8f D = __builtin_amdgcn_wmma_f32_16x16x4_f32(false, A, false, B, (short)0, C, false, false);
    float s = D[0]+D[1]+D[2]+D[3]+D[4]+D[5]+D[6]+D[7];
    s += __shfl_xor(s, 16, 32);
    return s;   // every lane holds the full wave sum
}

__global__ void rl_zero_kernel(float* p) { p[0] = 0.0f; }

// ---------------------------------------------------------------------------
// Kernel 1: per-GT-box preprocessing. Grid: NB blocks x 64 threads.
// Record layout (floats unless noted):
//   [0]=gl [1]=gr [2]=gt [3]=gb      (precomputed GT box edges, grid units)
//   [4]=SIL_FRAC*garea               (threshold partial, no div in hot loop)
//   [5]=match code as int bits: (best_a<<12)|(gj<<6)|gi, or -1 if invalid
//   [6]=tx [7]=ty [8]=tw [9]=th [10]=tconf [11]=tcls
// ---------------------------------------------------------------------------
__global__ void rl_prep_kernel(const float* __restrict__ out,
                               const float* __restrict__ target,
                               const float* __restrict__ anchors,
                               float* __restrict__ gtbl) {
    int b = blockIdx.x;
    int t = threadIdx.x;
    if (t >= TBOX) return;
    const float* tg = target + (b * TBOX + t) * 5;
    float cls = tg[0], gxn = tg[1], gyn = tg[2], gwn = tg[3], ghn = tg[4];
    bool valid = gxn > 0.0f;
    float gx = gxn * NW, gy = gyn * NH, gw = gwn * NW, gh = ghn * NH;

    // argmax over anchor IoUs (first max wins, like jnp.argmax)
    int best = 0; float bestiou = -1.0f;
    for (int n = 0; n < NA; ++n) {
        float aw = anchors[2*n], ah = anchors[2*n+1];
        float inter = fminf(gw, aw) * fminf(gh, ah);
        float uni = fmaxf(gw*gh + aw*ah - inter, 1e-10f);
        float iou = inter / uni;
        if (iou > bestiou) { bestiou = iou; best = n; }
    }
    int gi = (int)floorf(gx); gi = min(max(gi, 0), NW - 1);
    int gj = (int)floorf(gy); gj = min(max(gj, 0), NH - 1);
    float aw = anchors[2*best], ah = anchors[2*best+1];

    // predicted box at the target cell (precise math: tconf enters the loss)
    long base = ((long)(b * NA + best) * (5 + NC)) * PLANE + (long)gj * NW + gi;
    float o0 = out[base];
    float o1 = out[base + 1l*PLANE];
    float o2 = out[base + 2l*PLANE];
    float o3 = out[base + 3l*PLANE];
    float px = 1.0f / (1.0f + expf(-o0)) + (float)gi;
    float py = 1.0f / (1.0f + expf(-o1)) + (float)gj;
    float pw = expf(o2) * aw;
    float ph = expf(o3) * ah;

    float l  = fmaxf(gx - gw*0.5f, px - pw*0.5f);
    float r  = fminf(gx + gw*0.5f, px + pw*0.5f);
    float tt = fmaxf(gy - gh*0.5f, py - ph*0.5f);
    float bb = fminf(gy + gh*0.5f, py + ph*0.5f);
    float inter = fmaxf(r - l, 0.0f) * fmaxf(bb - tt, 0.0f);
    float uni = fmaxf(gw*gh + pw*ph - inter, 1e-10f);
    float iou_gt = inter / uni;

    float* rec = gtbl + (b * TBOX + t) * REC;
    rec[0]  = gx - gw * 0.5f;
    rec[1]  = gx + gw * 0.5f;
    rec[2]  = gy - gh * 0.5f;
    rec[3]  = gy + gh * 0.5f;
    rec[4]  = SIL_FRAC * (gw * gh);
    int code = valid ? ((best << 12) | (gj << 6) | gi) : -1;
    rec[5]  = __int_as_float(code);
    rec[6]  = gx - (float)gi;
    rec[7]  = gy - (float)gj;
    rec[8]  = logf(fmaxf(gw, 1e-10f) / aw);
    rec[9]  = logf(fmaxf(gh, 1e-10f) / ah);
    rec[10] = iou_gt;
    rec[11] = cls;
    rec[12] = 0.0f; rec[13] = 0.0f; rec[14] = 0.0f; rec[15] = 0.0f;
}

// ---------------------------------------------------------------------------
// Kernel 2: per-cell loss. 2560 blocks x 256 threads (8 wave32s).
// Each block = 256 contiguous cells (1/16 of one (b,a) plane), so b and a are
// derived from blockIdx.x only -> provably wave-uniform -> SALU/scalar loads.
// Hot loop: no division, no branches (invalid boxes give inter=0 < threshold
// and code=-1 which never matches). Class logits read only at matched cells.
// ---------------------------------------------------------------------------
__global__ __launch_bounds__(256) void rl_loss_kernel(const float* __restrict__ out,
                                                      const float* __restrict__ anchors,
                                                      const float* __restrict__ gtbl,
                                                      float* __restrict__ partial_out) {
    __shared__ float sg[TBOX * REC];
    __shared__ float swarp[8];
    int tid = threadIdx.x;
    // scalar (block-uniform) indices
    int b       = blockIdx.x / (NA * BLKS_PER_PLANE);
    int rem_blk = blockIdx.x % (NA * BLKS_PER_PLANE);
    int a       = rem_blk / BLKS_PER_PLANE;
    int pos0    = (rem_blk % BLKS_PER_PLANE) * 256;   // block's plane offset
    int pos = pos0 + tid;
    int j = pos / NW, i = pos % NW;

    const float* gsrc = gtbl + b * (TBOX * REC);      // scalar base
    for (int idx = tid; idx < TBOX * REC; idx += 256)
        sg[idx] = gsrc[idx];
    __syncthreads();

    long base = ((long)(b * NA + a) * (5 + NC)) * PLANE + pos;
    float o0 = out[base];
    float o1 = out[base + 1l*PLANE];
    float o2 = out[base + 2l*PLANE];
    float o3 = out[base + 3l*PLANE];
    float o4 = out[base + 4l*PLANE];

    float x = sigmoidf_(o0), y = sigmoidf_(o1);
    float w = o2, h = o3;
    float conf = sigmoidf_(o4);
    float aw = anchors[2*a], ah = anchors[2*a+1];     // scalar loads (a uniform)
    float px = x + (float)i, py = y + (float)j;
    float pw = __expf(w) * aw, ph = __expf(h) * ah;
    float pl = px - pw*0.5f, pr = px + pw*0.5f;
    float pt = py - ph*0.5f, pb = py + ph*0.5f;
    float pthr = SIL_FRAC * (pw * ph);                // per-thread threshold partial

    bool over_sil = false;
    int match = -1;                                   // last matching GT wins
    int mycode = (a << 12) | (j << 6) | i;
    #pragma unroll 5
    for (int t = 0; t < TBOX; ++t) {
        const float* rec = &sg[t * REC];
        float l  = fmaxf(rec[0], pl);
        float r  = fminf(rec[1], pr);
        float tt = fmaxf(rec[2], pt);
        float bb = fminf(rec[3], pb);
        float inter = fmaxf(r - l, 0.0f) * fmaxf(bb - tt, 0.0f);
        over_sil = over_sil || (inter > rec[4] + pthr);   // iou > 0.6, div-free
        if (__float_as_int(rec[5]) == mycode) match = t;
    }

    float conf_mask = over_sil ? 0.0f : 1.0f;   // NOOBJ_SCALE = 1
    float tx = 0.5f, ty = 0.5f, tw = 0.0f, th = 0.0f, tconf = 0.0f;
    float loss = 0.0f;

    if (match >= 0) {
        const float* rec = &sg[match * REC];
        tx = rec[6]; ty = rec[7]; tw = rec[8]; th = rec[9]; tconf = rec[10];
        conf_mask = OBJ_SCALE;
        int tc = (int)rec[11];
        float logits[NC];
        float m = -1e30f;
        #pragma unroll
        for (int c = 0; c < NC; ++c) {
            logits[c] = out[base + (long)(5 + c) * PLANE];
            m = fmaxf(m, logits[c]);
        }
        float s = 0.0f;
        #pragma unroll
        for (int c = 0; c < NC; ++c) s += __expf(logits[c] - m);
        float lse = m + __logf(s);
        loss += -(logits[tc] - lse);            // CLASS_SCALE = 1
    }

    float dx = x - tx, dy = y - ty, dw = w - tw, dh = h - th, dc = conf - tconf;
    loss += 0.5f * (dx*dx + dy*dy + dw*dw + dh*dh)    // COORD_SCALE = 1
          + 0.5f * conf_mask * dc * dc;

    // full-wave reduction via WMMA (EXEC all-1s: full block, reconverged)
    float wsum = wave32_sum_wmma(loss);
    int lane = tid & 31, wid = tid >> 5;
    if (lane == 0) swarp[wid] = wsum;
    __syncthreads();
    if (tid == 0) {
        float tot = 0.0f;
        #pragma unroll
        for (int k = 0; k < 8; ++k) tot += swarp[k];
        partial_out[blockIdx.x] = tot;
    }
}

// ---------------------------------------------------------------------------
// Kernel 3: deterministic final reduction of block partials (1 block).
// ---------------------------------------------------------------------------
__global__ __launch_bounds__(256) void rl_reduce_kernel(const float* __restrict__ part,
                                                        int n, float* __restrict__ d_out) {
    __shared__ float swarp[8];
    int tid = threadIdx.x;
    float s = 0.0f;
    for (int k = tid; k < n; k += 256) s += part[k];
    float wsum = wave32_sum_wmma(s);
    int lane = tid & 31, wid = tid >> 5;
    if (lane == 0) swarp[wid] = wsum;
    __syncthreads();
    if (tid == 0) {
        float tot = 0.0f;
        #pragma unroll
        for (int k = 0; k < 8; ++k) tot += swarp[k];
        d_out[0] = tot;
    }
}

extern "C" void kernel_launch(void* const* d_in, const int* in_sizes, int n_in,
                              void* d_out, int out_size, void* d_ws, size_t ws_size,
                              hipStream_t stream) {
    const float* output  = (const float*)d_in[0];  // (32,125,64,64) f32
    const float* target  = (const float*)d_in[1];  // (32,250) f32
    const float* anchors = (const float*)d_in[2];  // (10,) f32
    float* loss = (float*)d_out;

    // workspace layout: [GT table: NB*TBOX*REC floats][block partials]
    float* gtbl = (float*)d_ws;
    const int n_gt_floats = NB * TBOX * REC;                 // 25,600 floats
    const long N = (long)NB * NA * NH * NW;                  // 655,360 cells
    const int nblocks = (int)(N / 256);                      // 2,560
    float* partials = gtbl + n_gt_floats;

    rl_zero_kernel<<<1, 1, 0, stream>>>(loss);
    rl_prep_kernel<<<NB, 64, 0, stream>>>(output, target, anchors, gtbl);
    rl_loss_kernel<<<nblocks, 256, 0, stream>>>(output, anchors, gtbl, partials);
    rl_reduce_kernel<<<1, 256, 0, stream>>>(partials, nblocks, loss);
}